// EnhancedAdaptiveLoRAPooling_154618822919
// MI455X (gfx1250) — compile-verified
//
#include <hip/hip_runtime.h>
#include <math.h>

typedef float v2f __attribute__((ext_vector_type(2)));
typedef float v8f __attribute__((ext_vector_type(8)));

#define H 768
#define NTASK 16
#define RK 8
#define PITCH_A 772   // 16 x H Aeff in LDS, pitch % 64 == 4 -> conflict free
#define PITCH_B 17    // H x 16 Beff in LDS, gcd(17,64)==1 -> conflict free
#define PITCH_T 18    // 16 x 16 t-tile, even pitch keeps 8B align, distinct banks

// ---------------------------------------------------------------------------
// Kernel 1: per-task similarity (cos + euclid + 4-layer MLP). 16 blocks.
// ---------------------------------------------------------------------------
__global__ void __launch_bounds__(256) sims_kernel(
    const float* __restrict__ cur, const float* __restrict__ embs,
    const float* __restrict__ W1, const float* __restrict__ b1,
    const float* __restrict__ W2, const float* __restrict__ b2,
    const float* __restrict__ W3, const float* __restrict__ b3,
    const float* __restrict__ W4, const float* __restrict__ b4,
    float* __restrict__ sims) {
  __shared__ float z[2 * H];     // [cur, emb_n]
  __shared__ float h1[512];
  __shared__ float h2[256];
  __shared__ float h3[128];
  __shared__ float red[256];
  __shared__ float scal[4];      // dot(e,c), dot(e,e), dot(c,c), dot(d,d)

  const int n = blockIdx.x;
  const int t = threadIdx.x;
  const float* e = embs + n * H;

  for (int i = t; i < H; i += 256) { z[i] = cur[i]; z[H + i] = e[i]; }
  __syncthreads();

  float dec = 0.f, dee = 0.f, dcc = 0.f, ddd = 0.f;
  for (int i = t; i < H; i += 256) {
    float c = z[i], ev = z[H + i];
    dec += ev * c; dee += ev * ev; dcc += c * c;
    float df = ev - c; ddd += df * df;
  }
  float vals[4] = {dec, dee, dcc, ddd};
  for (int q = 0; q < 4; ++q) {
    red[t] = vals[q];
    __syncthreads();
    for (int off = 128; off > 0; off >>= 1) {
      if (t < off) red[t] += red[t + off];
      __syncthreads();
    }
    if (t == 0) scal[q] = red[0];
    __syncthreads();
  }

  // MLP: 1536 -> 512 -> 256 -> 128 -> 1
  for (int j = t; j < 512; j += 256) {
    float s = b1[j];
    const float* w = W1 + (size_t)j * (2 * H);
    for (int i = 0; i < 2 * H; ++i) s += w[i] * z[i];
    h1[j] = fmaxf(s, 0.f);
  }
  __syncthreads();
  {
    int j = t;  // 256 outputs, 256 threads
    float s = b2[j];
    const float* w = W2 + (size_t)j * 512;
    for (int i = 0; i < 512; ++i) s += w[i] * h1[i];
    h2[j] = fmaxf(s, 0.f);
  }
  __syncthreads();
  if (t < 128) {
    float s = b3[t];
    const float* w = W3 + (size_t)t * 256;
    for (int i = 0; i < 256; ++i) s += w[i] * h2[i];
    h3[t] = fmaxf(s, 0.f);
  }
  __syncthreads();
  if (t == 0) {
    float s = b4[0];
    for (int i = 0; i < 128; ++i) s += W4[i] * h3[i];
    float nn = 1.f / (1.f + expf(-s));
    float coss = scal[0] / fmaxf(sqrtf(scal[1]) * sqrtf(scal[2]), 1e-8f);
    float euclid = sqrtf(scal[3]);
    sims[n] = 0.4f * coss + 0.3f * (1.f / (1.f + euclid)) + 0.3f * nn;
  }
}

// ---------------------------------------------------------------------------
// Kernel 2: top-3, weights, fused effective rank-16 LoRA params. 1 block.
//   Aeff[16][H] : rows 0..7 = A_cur, rows 8..15 = pooled_a / safe_w
//   Beff[H][16] : cols 0..7 = c1 * B_cur, cols 8..15 = c2 * pooled_b / safe_w
// ---------------------------------------------------------------------------
__global__ void __launch_bounds__(256) combine_kernel(
    const float* __restrict__ cur,
    const float* __restrict__ loras_a, const float* __restrict__ loras_b,
    const int* __restrict__ task_id_p, const float* __restrict__ sims,
    float* __restrict__ Aeff, float* __restrict__ Beff) {
  __shared__ float red[256];
  __shared__ float sv[3];
  __shared__ int   sidx[3];
  __shared__ float sc[3];  // c1, c2, 1/safe_w

  const int t = threadIdx.x;
  float s = 0.f;
  for (int i = t; i < H; i += 256) { float c = cur[i]; s += c * c; }
  red[t] = s;
  __syncthreads();
  for (int off = 128; off > 0; off >>= 1) {
    if (t < off) red[t] += red[t + off];
    __syncthreads();
  }
  if (t == 0) {
    float cur_norm = sqrtf(red[0]);
    float v[NTASK];
    for (int i = 0; i < NTASK; ++i) v[i] = sims[i];
    float tw = 0.f;
    for (int k = 0; k < 3; ++k) {   // top-3, first-occurrence tie-break
      int bi = 0; float bv = v[0];
      for (int i = 1; i < NTASK; ++i) { if (v[i] > bv) { bv = v[i]; bi = i; } }
      float w = (bv > 0.f) ? bv : 0.f;
      sidx[k] = bi; sv[k] = w; tw += w;
      v[bi] = -1e30f;
    }
    float safe = (tw > 0.f) ? tw : 1.f;
    float fw = fminf(cur_norm * 0.1f, 0.5f);
    sc[0] = 2.f * ((tw > 0.f) ? (1.f - fw) : 1.f);  // c1
    sc[1] = (tw > 0.f) ? (2.f * fw) : 0.f;          // c2
    sc[2] = 1.f / safe;
  }
  __syncthreads();

  const int cid = *task_id_p;
  const float c1 = sc[0], c2 = sc[1], invw = sc[2];
  const float w0 = sv[0] * invw, w1 = sv[1] * invw, w2 = sv[2] * invw;
  const int i0 = sidx[0], i1 = sidx[1], i2 = sidx[2];

  // Aeff: [16][H] row-major
  for (int idx = t; idx < RK * H; idx += 256) {
    Aeff[idx] = loras_a[(size_t)cid * RK * H + idx];
    Aeff[RK * H + idx] = w0 * loras_a[(size_t)i0 * RK * H + idx]
                       + w1 * loras_a[(size_t)i1 * RK * H + idx]
                       + w2 * loras_a[(size_t)i2 * RK * H + idx];
  }
  // Beff: [H][16] row-major (loras_b is [N][H][R])
  for (int idx = t; idx < H * RK; idx += 256) {
    int h = idx >> 3, c = idx & 7;
    Beff[h * 16 + c] = c1 * loras_b[(size_t)cid * H * RK + idx];
    Beff[h * 16 + 8 + c] =
        c2 * (w0 * loras_b[(size_t)i0 * H * RK + idx]
            + w1 * loras_b[(size_t)i1 * H * RK + idx]
            + w2 * loras_b[(size_t)i2 * H * RK + idx]);
  }
}

// ---------------------------------------------------------------------------
// Kernel 3: streaming pass. out = x + (x @ Aeff^T) @ Beff^T.
// One wave per 16-row tile, 8 waves (256 threads) per block.
// Stage 1: t[16x16] accumulated with V_WMMA_F32_16X16X4_F32 over K=768.
// Stage 2: 48 column chunks, each 4 WMMAs over K=16, + residual add.
// ---------------------------------------------------------------------------
__global__ void __launch_bounds__(256) main_kernel(
    const float* __restrict__ x,
    const float* __restrict__ Aeff, const float* __restrict__ Beff,
    float* __restrict__ out) {
  __shared__ float sA[16 * PITCH_A];      // 49.4 KB
  __shared__ float sB[H * PITCH_B];       // 52.2 KB
  __shared__ float sT[8 * 16 * PITCH_T];  //  9.2 KB

  const int t = threadIdx.x;
  for (int i = t; i < 16 * H; i += 256) {
    int r = i / H, c = i - r * H;
    sA[r * PITCH_A + c] = Aeff[i];
  }
  for (int i = t; i < H * 16; i += 256) {
    int r = i >> 4, c = i & 15;
    sB[r * PITCH_B + c] = Beff[i];
  }
  __syncthreads();

  const int lane = t & 31;
  const int wave = t >> 5;
  const int n = lane & 15;               // M for A-frags, N for B/D
  const int kk = (lane >> 4) << 1;       // 0 or 2 (K sub-pair)
  const int mhalf = (lane >> 4) * 8;     // D-layout row half

  const size_t tile = (size_t)blockIdx.x * 8 + wave;
  const float* xb = x + tile * (16 * H);
  float* ob = out + tile * (16 * H);

  // ---- stage 1: t = Xtile @ Aeff^T  (D[m][n] = sum_k X[m][k]*Aeff[n][k]) ----
  v8f acc = {0.f, 0.f, 0.f, 0.f, 0.f, 0.f, 0.f, 0.f};
  for (int k0 = 0; k0 < H; k0 += 4) {
    const v2f a = *(const v2f*)(xb + n * H + k0 + kk);   // 8B-aligned (k0+kk even)
    v2f b;
    b.x = sA[n * PITCH_A + k0 + kk];
    b.y = sA[n * PITCH_A + k0 + kk + 1];
    acc = __builtin_amdgcn_wmma_f32_16x16x4_f32(false, a, false, b,
                                                (short)0, acc, false, false);
  }

  // spill t (D layout) to LDS so we can re-read it in A layout (same wave,
  // DS ops are in-order per wave -> no barrier needed)
  float* tp = &sT[wave * (16 * PITCH_T)];
#pragma unroll
  for (int v = 0; v < 8; ++v) tp[(mhalf + v) * PITCH_T + n] = acc[v];

  // ---- stage 2: out = x + t @ Beff^T, 16-column chunks ----
  for (int c0 = 0; c0 < H; c0 += 16) {
    v8f d = {0.f, 0.f, 0.f, 0.f, 0.f, 0.f, 0.f, 0.f};
#pragma unroll
    for (int k0 = 0; k0 < 16; k0 += 4) {
      v2f a, b;
      a.x = tp[n * PITCH_T + k0 + kk];
      a.y = tp[n * PITCH_T + k0 + kk + 1];
      b.x = sB[(c0 + n) * PITCH_B + k0 + kk];
      b.y = sB[(c0 + n) * PITCH_B + k0 + kk + 1];
      d = __builtin_amdgcn_wmma_f32_16x16x4_f32(false, a, false, b,
                                                (short)0, d, false, false);
    }
#pragma unroll
    for (int v = 0; v < 8; ++v) {
      int m = mhalf + v;
      int col = c0 + n;
      ob[m * H + col] = xb[m * H + col] + d[v];
    }
  }
}

// ---------------------------------------------------------------------------
extern "C" void kernel_launch(void* const* d_in, const int* in_sizes, int n_in,
                              void* d_out, int out_size, void* d_ws, size_t ws_size,
                              hipStream_t stream) {
  const float* x    = (const float*)d_in[0];
  const float* cur  = (const float*)d_in[1];
  const float* la   = (const float*)d_in[2];
  const float* lb   = (const float*)d_in[3];
  const float* embs = (const float*)d_in[4];
  const float* W1   = (const float*)d_in[5];
  const float* b1   = (const float*)d_in[6];
  const float* W2   = (const float*)d_in[7];
  const float* b2   = (const float*)d_in[8];
  const float* W3   = (const float*)d_in[9];
  const float* b3   = (const float*)d_in[10];
  const float* W4   = (const float*)d_in[11];
  const float* b4   = (const float*)d_in[12];
  const int*   tid  = (const int*)d_in[13];

  float* ws   = (float*)d_ws;
  float* sims = ws;
  float* Aeff = ws + 16;
  float* Beff = Aeff + 16 * H;
  float* out  = (float*)d_out;

  sims_kernel<<<NTASK, 256, 0, stream>>>(cur, embs, W1, b1, W2, b2, W3, b3,
                                         W4, b4, sims);
  combine_kernel<<<1, 256, 0, stream>>>(cur, la, lb, tid, sims, Aeff, Beff);
  // 32768 rows / 16 per tile = 2048 tiles; 8 tiles (waves) per block.
  main_kernel<<<256, 256, 0, stream>>>(x, Aeff, Beff, out);
}